// bpd_cuda_python_11235634446935
// MI455X (gfx1250) — compile-verified
//
#include <hip/hip_runtime.h>
#include <hip/hip_bf16.h>

#ifndef __has_builtin
#define __has_builtin(x) 0
#endif

#define HH 512
#define WW 512
#define NN (HH * WW)
#define PIF 3.14159265f
#define TWO_PIF (2.0f * PIF)
#define TILE_W 32
#define TILE_H 8
#define HALO_W (TILE_W + 2)
#define HALO_H (TILE_H + 2)
#define CC_ITERS 48

#if defined(__HIP_DEVICE_COMPILE__) && __has_builtin(__builtin_amdgcn_tensor_load_to_lds)
#define HAVE_TDM 1
#else
#define HAVE_TDM 0
#endif

typedef unsigned int u32x4 __attribute__((ext_vector_type(4)));
typedef int i32x4 __attribute__((ext_vector_type(4)));
typedef int i32x8 __attribute__((ext_vector_type(8)));

__device__ __forceinline__ void wait_tensorcnt0() {
#if __has_builtin(__builtin_amdgcn_s_wait_tensorcnt)
  __builtin_amdgcn_s_wait_tensorcnt(0);
#else
  asm volatile("s_wait_tensorcnt 0x0" ::: "memory");
#endif
}

// Pinned LDS read: ds_load_b32 from a raw LDS byte offset (low 32 bits of the
// flat shared address per the aperture mapping), completed with s_wait_dscnt.
__device__ __forceinline__ float lds_load_f32(unsigned int byte_off) {
  float v;
  asm volatile("ds_load_b32 %0, %1\n\ts_wait_dscnt 0x0"
               : "=v"(v)
               : "v"(byte_off));
  return v;
}

// angle -> direction bin -> (dh, dw) via packed nibble tables
__device__ __forceinline__ void dir_of(float a, int& dh, int& dw) {
  float p = rintf((a + PIF * 0.125f) / (PIF * 0.25f));
  if (p >= 8.0f) p -= 8.0f;
  int pos = (int)p;
  dh = (int)((0x21000122u >> (4 * pos)) & 0xFu) - 1;
  dw = (int)((0x00012221u >> (4 * pos)) & 0xFu) - 1;
}

// ---------------------------------------------------------------------------
// K1: per-pixel parent pointer. Interior 32x8 tiles DMA their 34x10 halo tile
// into LDS with the Tensor Data Mover (one tensor_load_to_lds per block,
// issued by wave 0, tracked by TENSORcnt), then read it back with explicit
// ds_load_b32. Border tiles use clamped global loads (TDM OOB reads return
// zero, which would not match the clamp semantics).
// ---------------------------------------------------------------------------
__global__ void k_parents(const float* __restrict__ ang,
                          const int* __restrict__ theta_p,
                          int* __restrict__ pf) {
  __shared__ float tile[HALO_H * HALO_W];

  const int tx = threadIdx.x, ty = threadIdx.y;
  const int c0 = blockIdx.x * TILE_W;
  const int r0 = blockIdx.y * TILE_H;
  const bool interior = (blockIdx.x > 0) && (blockIdx.y > 0) &&
                        (c0 + TILE_W + 1 <= WW) && (r0 + TILE_H + 1 <= HH);

  const int r = r0 + ty, c = c0 + tx;
  const float th = (float)(*theta_p) * (PIF / 180.0f);
  int result;

  if (interior) {
    unsigned int lds_base = (unsigned int)(unsigned long long)(void*)&tile[0];
#if HAVE_TDM
    if (ty == 0) {  // one wave32 issues the DMA; EXEC-safe (uniform branch)
      unsigned long long ga =
          (unsigned long long)(const void*)(ang + (size_t)(r0 - 1) * WW + (c0 - 1));
      // D# group0: count=1 | lds_addr | global_addr[56:0] | type=2
      u32x4 g0 = {1u, lds_base, (unsigned int)ga,
                  (((unsigned int)(ga >> 32)) & 0x01FFFFFFu) | (2u << 30)};
      // D# group1: data_size=4B, tensor 512x512, tile 34x10, dim0_stride=512
      i32x8 g1 = {(int)(2u << 16),
                  (int)(((unsigned)WW & 0xFFFFu) << 16),
                  (int)((((unsigned)WW >> 16) & 0xFFFFu) |
                        (((unsigned)HH & 0xFFFFu) << 16)),
                  (int)((((unsigned)HH >> 16) & 0xFFFFu) |
                        ((unsigned)HALO_W << 16)),
                  (int)HALO_H,
                  (int)WW,
                  0, 0};
      i32x4 z4 = {0, 0, 0, 0};
#if __clang_major__ >= 23
      i32x8 z8 = {0, 0, 0, 0, 0, 0, 0, 0};
      __builtin_amdgcn_tensor_load_to_lds(g0, g1, z4, z4, z8, 0);
#else
      __builtin_amdgcn_tensor_load_to_lds(g0, g1, z4, z4, 0);
#endif
      wait_tensorcnt0();
    }
#else
    // Fallback: cooperative LDS fill
    int tid = ty * TILE_W + tx;
    for (int k = tid; k < HALO_H * HALO_W; k += TILE_H * TILE_W) {
      int rr = k / HALO_W, cc = k - rr * HALO_W;
      tile[k] = ang[(size_t)(r0 - 1 + rr) * WW + (c0 - 1 + cc)];
    }
#endif
    __syncthreads();

    // explicit ds_load_b32 reads of the DMA-filled tile
    float a = lds_load_f32(lds_base +
                           4u * (unsigned)((ty + 1) * HALO_W + (tx + 1)));
    int dh, dw;
    dir_of(a, dh, dw);
    float an = lds_load_f32(
        lds_base + 4u * (unsigned)((ty + 1 + dh) * HALO_W + (tx + 1 + dw)));
    float ad = fabsf(a - an);
    ad = fminf(ad, TWO_PIF - ad);
    // interior => neighbor always in bounds
    result = (ad > th) ? (r * WW + c) : ((r + dh) * WW + (c + dw));
  } else {
    float a = ang[r * WW + c];
    int dh, dw;
    dir_of(a, dh, dw);
    int nh = r + dh, nw = c + dw;
    bool inb = (nh >= 0) && (nh < HH) && (nw >= 0) && (nw < WW);
    int nhc = min(max(nh, 0), HH - 1), nwc = min(max(nw, 0), WW - 1);
    float an = ang[nhc * WW + nwc];
    float ad = fabsf(a - an);
    ad = fminf(ad, TWO_PIF - ad);
    bool root = (!inb) || (ad > th);
    result = root ? (r * WW + c) : (nhc * WW + nwc);
  }
  pf[r * WW + c] = result;
}

// ---------------------------------------------------------------------------
// K2: init labels + precompute the 18-bit root-merge window mask once
// (roots are invariant across all 48 CC rounds).
// ---------------------------------------------------------------------------
__global__ void k_init(const int* __restrict__ pf, int* __restrict__ f,
                       unsigned int* __restrict__ emask) {
  int i = blockIdx.x * blockDim.x + threadIdx.x;
  if (i >= NN) return;
  f[i] = i;
  int r = i >> 9, c = i & (WW - 1);
  unsigned int m = 0;
  if (pf[i] == i) {
    int bit = 0;
    for (int dh = 0; dh <= 2; ++dh) {
      for (int dw = -3; dw <= 2; ++dw, ++bit) {
        int nr = r + dh, nc = c + dw;
        if (nr <= HH - 2 && nc >= 0 && nc <= WW - 2) {
          int v = nr * WW + nc;
          if (pf[v] == v) m |= (1u << bit);
        }
      }
    }
    m &= ~(1u << 3);  // drop the (0,0) self edge
  }
  emask[i] = m;
}

__device__ __forceinline__ void hook_edge(int* f, int u, int v) {
  int fu = f[u], fv = f[v];
  if (fu == fv) return;
  int mn = fu < fv ? fu : fv;
  atomicMin(&f[fu], mn);
  atomicMin(&f[fv], mn);
  atomicMin(&f[u], mn);
  atomicMin(&f[v], mn);
}

// K3: FastSV hook round (in-place monotone scatter-min; same fixed point).
__global__ void k_hook(const int* __restrict__ pf,
                       const unsigned int* __restrict__ emask, int* f) {
  int i = blockIdx.x * blockDim.x + threadIdx.x;
  if (i >= NN) return;
  hook_edge(f, i, pf[i]);
  unsigned int m = emask[i];
  while (m) {
    int b = __ffs(m) - 1;
    m &= m - 1;
    int v = i + (b / 6) * WW + (b % 6) - 3;
    hook_edge(f, i, v);
  }
}

// K4: pointer jumping (multi-step chase; monotone, plain store is safe).
__global__ void k_jump(int* __restrict__ f) {
  int i = blockIdx.x * blockDim.x + threadIdx.x;
  if (i >= NN) return;
  __builtin_prefetch(&f[(i + 8192) & (NN - 1)], 0, 1);  // global_prefetch_b8
  int v = f[i];
  v = f[v];
  v = f[v];
  v = f[v];
  if (v < f[i]) f[i] = v;
}

// K5: 3-step ancestor chase.
__global__ void k_c3(const int* __restrict__ pf, int* __restrict__ c3) {
  int i = blockIdx.x * blockDim.x + threadIdx.x;
  if (i >= NN) return;
  int p = pf[i];
  p = pf[p];
  p = pf[p];
  c3[i] = p;
}

// K6: boundary wrap-around angle diff for right/down neighbors.
__global__ void k_out(const float* __restrict__ ang, const int* __restrict__ c3,
                      const int* __restrict__ f, float* __restrict__ out) {
  int i = blockIdx.x * blockDim.x + threadIdx.x;
  if (i >= NN) return;
  int r = i >> 9, c = i & (WW - 1);
  float a0 = ang[c3[i]];
  int li = f[i];
  float o0 = 0.0f, o1 = 0.0f;
  if (c < WW - 1) {
    int n = i + 1;
    if (f[n] != li) {
      float d = fabsf(a0 - ang[c3[n]]);
      o0 = fminf(d, TWO_PIF - d);
    }
  }
  if (r < HH - 1) {
    int n = i + WW;
    if (f[n] != li) {
      float d = fabsf(a0 - ang[c3[n]]);
      o1 = fminf(d, TWO_PIF - d);
    }
  }
  out[2 * i] = o0;
  out[2 * i + 1] = o1;
}

extern "C" void kernel_launch(void* const* d_in, const int* in_sizes, int n_in,
                              void* d_out, int out_size, void* d_ws,
                              size_t ws_size, hipStream_t stream) {
  (void)in_sizes; (void)n_in; (void)out_size; (void)ws_size;
  const float* ang = (const float*)d_in[0];
  const int* theta_p = (const int*)d_in[3];  // height/width fixed at 512 by setup

  char* ws = (char*)d_ws;
  int* pf = (int*)(ws);                                  // 1 MB
  unsigned int* emask = (unsigned int*)(ws + (size_t)NN * 4);   // 1 MB
  int* f = (int*)(ws + (size_t)NN * 8);                  // 1 MB
  int* c3 = (int*)(ws + (size_t)NN * 12);                // 1 MB
  float* out = (float*)d_out;

  dim3 blk(TILE_W, TILE_H), grd(WW / TILE_W, HH / TILE_H);
  k_parents<<<grd, blk, 0, stream>>>(ang, theta_p, pf);

  const int T = 256, B = NN / T;
  k_init<<<B, T, 0, stream>>>(pf, f, emask);
  for (int it = 0; it < CC_ITERS; ++it) {
    k_hook<<<B, T, 0, stream>>>(pf, emask, f);
    k_jump<<<B, T, 0, stream>>>(f);
  }
  k_c3<<<B, T, 0, stream>>>(pf, c3);
  k_out<<<B, T, 0, stream>>>(ang, c3, f, out);
}